// SequenceLevelHybridLoss_51771535786460
// MI455X (gfx1250) — compile-verified
//
#include <hip/hip_runtime.h>

// MI455X / gfx1250, wave32. Memory-bound problem: stream each input tensor
// from HBM once (stage A: WMMA logits), re-read the 1MB workgroup tile from
// the 192MB L2 (stage B: weighted accumulation). f32 WMMA keeps precision.

typedef __attribute__((ext_vector_type(2))) float v2f;
typedef __attribute__((ext_vector_type(8))) float v8f;

constexpr int   B_      = 16;
constexpr int   S_      = 2048;
constexpr int   HS_     = 2048;
constexpr int   HT_     = 1024;
constexpr int   TPB     = 256;            // 8 waves of 32
constexpr int   CHUNK   = 128;            // s-rows per workgroup (8 waves * 16)
constexpr int   NCHUNK  = S_ / CHUNK;     // 16
constexpr float NEG_INF_ = -1e9f;
constexpr float LAMBDA_  = 0.5f;
constexpr float COS_EPS_ = 1e-8f;
constexpr int   D_       = (HS_ < HT_ ? HS_ : HT_);   // compared dims (see note)

// ---------------- kernel 0: left-aligned mask -> integer lengths ------------
__global__ __launch_bounds__(TPB) void mask_len_kernel(
    const float* __restrict__ smask, const float* __restrict__ tmask,
    int* __restrict__ lens)
{
  __shared__ float red[TPB];
  const int idx = blockIdx.x;  // 0..15 student rows, 16..31 teacher rows
  const float* m = (idx < B_) ? (smask + (size_t)idx * S_)
                              : (tmask + (size_t)(idx - B_) * S_);
  float s = 0.f;
  for (int i = threadIdx.x; i < S_; i += TPB) s += m[i];
  red[threadIdx.x] = s;
  __syncthreads();
  for (int off = TPB / 2; off > 0; off >>= 1) {
    if (threadIdx.x < off) red[threadIdx.x] += red[threadIdx.x + off];
    __syncthreads();
  }
  if (threadIdx.x == 0) lens[idx] = (int)(red[0] + 0.5f);
}

// ---------------- kernel 1: per-(b,chunk) logits + softmax partials ---------
// Stage A: logits via V_WMMA_F32_16X16X4_F32 with w broadcast down the M rows:
//   C[m][n] = sum_k w[k] * x[row n][k]  -> every lane's c[0] is its row's dot.
// A 16x4 f32 layout: v0 = K0(l0-15)/K2(l16-31), v1 = K1/K3  -> per-lane float2
// of w at k + 2*half.  B 4x16 layout mirrors it -> per-lane float2 of x row n.
// Stage B: chunk-local softmax (m_c, l_c) and partial pooled acc; the x tile
// (1MB) is still L2-resident from stage A.
template<int H>
__global__ __launch_bounds__(TPB) void pool_partial_kernel(
    const float* __restrict__ x,      // [B, S, H]
    const float* __restrict__ w,      // [H]
    const float* __restrict__ bias,   // [1]
    const int*   __restrict__ lens,   // [B]
    int extra_pad,                    // 0 student, 1 teacher (attend-padding)
    float* __restrict__ part,         // [B, NCHUNK, H]
    float* __restrict__ ml)           // [B, NCHUNK, 2] = {m_c, l_c}
{
  __shared__ float s_logit[CHUNK];
  __shared__ float s_p[CHUNK];

  const int b     = blockIdx.y;
  const int chunk = blockIdx.x;
  const int s0    = chunk * CHUNK;
  const int tid   = threadIdx.x;
  const int wave  = tid >> 5;
  const int lane  = tid & 31;
  const int half  = lane >> 4;
  const int n     = lane & 15;
  const int lrow  = wave * 16 + n;        // row within chunk (B-matrix column)
  const int row   = s0 + lrow;
  const int limit = min(lens[b] + extra_pad, S_);

  const float* xb = x + (size_t)b * S_ * H;

  // ---- stage A: WMMA logits (EXEC all ones: no divergence here) ----
  const float* arow = w + 2 * half;                      // w broadcast fragment
  const float* brow = xb + (size_t)row * H + 2 * half;   // x data fragment
  v8f c = {0.f, 0.f, 0.f, 0.f, 0.f, 0.f, 0.f, 0.f};
#pragma unroll 4
  for (int k = 0; k < H; k += 4) {
    v2f a  = *(const v2f*)(arow + k);
    v2f bb = *(const v2f*)(brow + k);
    c = __builtin_amdgcn_wmma_f32_16x16x4_f32(false, a, false, bb,
                                              (short)0, c, false, false);
  }
  float logit = c[0] + bias[0];                 // every lane: dot of its row
  logit = (row < limit) ? logit : NEG_INF_;     // contiguous-mask rule
  if (half == 0) s_logit[lrow] = logit;
  __syncthreads();

  // ---- chunk-local softmax stats (exact: exp(-1e9 - m) underflows to 0) ----
  float m = -3.0e38f;
  for (int i = 0; i < CHUNK; ++i) m = fmaxf(m, s_logit[i]);
  if (tid < CHUNK) s_p[tid] = __expf(s_logit[tid] - m);
  __syncthreads();
  if (tid == 0) {
    float l = 0.f;
    for (int i = 0; i < CHUNK; ++i) l += s_p[i];
    ml[(b * NCHUNK + chunk) * 2 + 0] = m;
    ml[(b * NCHUNK + chunk) * 2 + 1] = l;
  }

  // ---- stage B: partial pooled accumulation (tile re-read from L2) ----
  constexpr int PH = H / TPB;                   // 8 (student) / 4 (teacher)
  const int h0 = tid * PH;
  float acc[PH];
#pragma unroll
  for (int j = 0; j < PH; ++j) acc[j] = 0.f;
  for (int s = 0; s < CHUNK; ++s) {
    if (s + 8 < CHUNK)                          // uniform; -> global_prefetch
      __builtin_prefetch(xb + (size_t)(s0 + s + 8) * H + h0, 0, 1);
    const float p = s_p[s];                     // LDS broadcast read
    const float4* xv = (const float4*)(xb + (size_t)(s0 + s) * H + h0);
#pragma unroll
    for (int j = 0; j < PH / 4; ++j) {
      float4 v = xv[j];
      acc[4 * j + 0] += p * v.x;
      acc[4 * j + 1] += p * v.y;
      acc[4 * j + 2] += p * v.z;
      acc[4 * j + 3] += p * v.w;
    }
  }
  float* dst = part + (size_t)(b * NCHUNK + chunk) * H + h0;
#pragma unroll
  for (int j = 0; j < PH; ++j) dst[j] = acc[j];
}

// ---------------- kernel 2: exact combine of softmax partials ---------------
template<int H>
__global__ __launch_bounds__(TPB) void pool_combine_kernel(
    const float* __restrict__ part, const float* __restrict__ ml,
    float* __restrict__ pooled)
{
  const int b = blockIdx.x;
  float M = -3.0e38f;
#pragma unroll
  for (int c = 0; c < NCHUNK; ++c) M = fmaxf(M, ml[(b * NCHUNK + c) * 2 + 0]);
  float coef[NCHUNK];
  float L = 0.f;
#pragma unroll
  for (int c = 0; c < NCHUNK; ++c) {
    float e = __expf(ml[(b * NCHUNK + c) * 2 + 0] - M);
    coef[c] = e;
    L += e * ml[(b * NCHUNK + c) * 2 + 1];
  }
  const float invL = 1.0f / L;
  for (int h = threadIdx.x; h < H; h += TPB) {
    float acc = 0.f;
#pragma unroll
    for (int c = 0; c < NCHUNK; ++c)
      acc += coef[c] * part[(size_t)(b * NCHUNK + c) * H + h];
    pooled[(size_t)b * H + h] = acc * invL;
  }
}

// ---------------- kernel 3: per-row loss statistics -------------------------
// NOTE: reference subtracts [B,2048] - [B,1024] (shape-invalid in JAX); we
// compare over the shared leading D_=1024 dims consistently (dot/norms too).
__global__ __launch_bounds__(TPB) void loss_rows_kernel(
    const float* __restrict__ sp, const float* __restrict__ tp,
    float* __restrict__ row_stats)   // [B][2] = {huber_sum, cos_term}
{
  __shared__ float4 red[TPB];
  const int b = blockIdx.x;
  const int tid = threadIdx.x;
  float hub = 0.f, dot = 0.f, na2 = 0.f, nb2 = 0.f;
  for (int h = tid; h < D_; h += TPB) {
    float a = sp[(size_t)b * HS_ + h];
    float t = tp[(size_t)b * HT_ + h];
    float d = a - t, ad = fabsf(d);
    hub += (ad <= 1.0f) ? 0.5f * d * d : (ad - 0.5f);   // delta = 1
    dot += a * t;
    na2 += a * a;
    nb2 += t * t;
  }
  red[tid] = make_float4(hub, dot, na2, nb2);
  __syncthreads();
  for (int off = TPB / 2; off > 0; off >>= 1) {
    if (tid < off) {
      float4 u = red[tid], v = red[tid + off];
      red[tid] = make_float4(u.x + v.x, u.y + v.y, u.z + v.z, u.w + v.w);
    }
    __syncthreads();
  }
  if (tid == 0) {
    float4 r = red[0];
    float na = fmaxf(sqrtf(r.z), COS_EPS_);
    float nb = fmaxf(sqrtf(r.w), COS_EPS_);
    row_stats[2 * b + 0] = r.x;
    row_stats[2 * b + 1] = 1.0f - r.y / (na * nb);
  }
}

// ---------------- kernel 4: final scalars -----------------------------------
__global__ void loss_final_kernel(const float* __restrict__ row_stats,
                                  float* __restrict__ out)
{
  if (threadIdx.x == 0 && blockIdx.x == 0) {
    float hub = 0.f, cosl = 0.f;
    for (int b = 0; b < B_; ++b) {
      hub  += row_stats[2 * b + 0];
      cosl += row_stats[2 * b + 1];
    }
    float huber = hub / (float)(B_ * D_);
    float cosm  = cosl / (float)B_;
    out[0] = (1.0f - LAMBDA_) * huber + LAMBDA_ * cosm;  // hybrid
    out[1] = cosm;
    out[2] = huber;
    out[3] = LAMBDA_;
  }
}

// ---------------- host launch ----------------------------------------------
extern "C" void kernel_launch(void* const* d_in, const int* in_sizes, int n_in,
                              void* d_out, int out_size, void* d_ws, size_t ws_size,
                              hipStream_t stream)
{
  (void)in_sizes; (void)n_in; (void)out_size; (void)ws_size;
  const float* sx    = (const float*)d_in[0];  // [B,S,HS]
  const float* tx    = (const float*)d_in[1];  // [B,S,HT]
  const float* smask = (const float*)d_in[2];  // [B,S]
  const float* tmask = (const float*)d_in[3];  // [B,S]
  const float* sw    = (const float*)d_in[4];  // [HS]
  const float* sb    = (const float*)d_in[5];  // [1]
  const float* tw    = (const float*)d_in[6];  // [HT]
  const float* tb    = (const float*)d_in[7];  // [1]
  float* out = (float*)d_out;

  char* ws = (char*)d_ws;
  size_t o = 0;
  int*   lens      = (int*)  (ws + o); o += 256;                               // 32 ints
  float* ml_s      = (float*)(ws + o); o += (size_t)B_ * NCHUNK * 2 * 4;       // 2KB
  float* ml_t      = (float*)(ws + o); o += (size_t)B_ * NCHUNK * 2 * 4;       // 2KB
  float* row_stats = (float*)(ws + o); o += 256;                               // 32 f
  float* pooled_s  = (float*)(ws + o); o += (size_t)B_ * HS_ * 4;              // 128KB
  float* pooled_t  = (float*)(ws + o); o += (size_t)B_ * HT_ * 4;              // 64KB
  float* part_s    = (float*)(ws + o); o += (size_t)B_ * NCHUNK * HS_ * 4;     // 2MB
  float* part_t    = (float*)(ws + o); o += (size_t)B_ * NCHUNK * HT_ * 4;     // 1MB

  mask_len_kernel<<<2 * B_, TPB, 0, stream>>>(smask, tmask, lens);

  pool_partial_kernel<HS_><<<dim3(NCHUNK, B_), TPB, 0, stream>>>(
      sx, sw, sb, lens, /*extra_pad=*/0, part_s, ml_s);
  pool_partial_kernel<HT_><<<dim3(NCHUNK, B_), TPB, 0, stream>>>(
      tx, tw, tb, lens + B_, /*extra_pad=*/1, part_t, ml_t);

  pool_combine_kernel<HS_><<<B_, TPB, 0, stream>>>(part_s, ml_s, pooled_s);
  pool_combine_kernel<HT_><<<B_, TPB, 0, stream>>>(part_t, ml_t, pooled_t);

  loss_rows_kernel<<<B_, TPB, 0, stream>>>(pooled_s, pooled_t, row_stats);
  loss_final_kernel<<<1, 32, 0, stream>>>(row_stats, out);
}